// DownSampleWithSigma_17145509446490
// MI455X (gfx1250) — compile-verified
//
#include <hip/hip_runtime.h>
#include <hip/hip_bf16.h>
#include <math.h>

// ---------------------------------------------------------------------------
// Problem constants (from reference): x (8,128,2048) f32; Wq/Wk/Wv (128,128).
// ---------------------------------------------------------------------------
#define BATCH 8
#define CH    128          // channels
#define NPTS  2048         // points
#define KNN   32           // neighbors
#define HEADS 4
#define DEPTH 32
#define TOPK  1024

typedef __attribute__((ext_vector_type(2))) float v2f;
typedef __attribute__((ext_vector_type(8))) float v8f;

// fp32 WMMA: D(16x16) += A(16x4) * B(4x16), 8-arg form
// (neg_a, A, neg_b, B, c_mod, C, reuse_a, reuse_b)
__device__ __forceinline__ v8f wmma_f32(v2f a, v2f b, v8f c) {
    return __builtin_amdgcn_wmma_f32_16x16x4_f32(false, a, false, b, (short)0, c,
                                                 false, false);
}

// Hard scheduling fence: nothing may be reordered across (mask = 0).
__device__ __forceinline__ void sched_fence() {
    __builtin_amdgcn_sched_barrier(0);
}

// Async DMA: global -> LDS, 16 bytes, tracked by ASYNCcnt (CDNA5).
// lds: byte address in LDS (low 32 bits of a generic LDS pointer).
__device__ __forceinline__ void async_g2l_b128(unsigned lds, const float* g) {
    asm volatile("global_load_async_to_lds_b128 %0, %1, off"
                 :: "v"(lds), "v"(g) : "memory");
}
__device__ __forceinline__ void wait_async0() {
    asm volatile("s_wait_asynccnt 0x0" ::: "memory");
}
__device__ __forceinline__ unsigned lds_addr(const void* p) {
    return (unsigned)(unsigned long long)p;   // LDS offset = addr[31:0]
}

// ---------------------------------------------------------------------------
// Kernel 1: sq[b][n] = sum_c x[b][c][n]^2
// ---------------------------------------------------------------------------
__global__ void sq_kernel(const float* __restrict__ x, float* __restrict__ sq) {
    int i = blockIdx.x * blockDim.x + threadIdx.x;       // 0 .. B*N-1
    int b = i / NPTS;
    int n = i % NPTS;
    const float* xb = x + (size_t)b * CH * NPTS;
    float s = 0.f;
    for (int c = 0; c < CH; ++c) {
        float v = xb[(size_t)c * NPTS + n];
        s += v * v;
    }
    sq[i] = s;
}

// ---------------------------------------------------------------------------
// Kernel 2: distances via fp32 WMMA + per-row top-32 (smallest) selection.
// One workgroup per (batch, 16-row block). 512 threads = 16 waves.
//
// Dynamic LDS (264 KB, only legal on CDNA5's 320 KB WGP LDS):
//   xqT  [CH][16]              8 KB   query tile (channel-major)
//   xc   [16 waves][CH][16]  128 KB   per-wave candidate tile
//   dist [16][NPTS]          128 KB   distance strip for top-k
//
// Staged with GLOBAL_LOAD_ASYNC_TO_LDS_B128 / s_wait_asynccnt.
// A operands (query tile) live in 64 VGPRs for the whole wave; B operands are
// pipelined in groups of 8 K-steps.  sched_barrier(0) pins the final order to
// [loads g+1][WMMAs g] so WMMAs wait on dscnt<=8..15 instead of 0.
// __launch_bounds__(512,1): the 264 KB LDS footprint limits residency to one
// WG per WGP regardless, so give the allocator the full VGPR budget (the
// ~150-VGPR live set must not spill to scratch).
// ---------------------------------------------------------------------------
__global__ void __launch_bounds__(512, 1)
knn_kernel(const float* __restrict__ x,
           const float* __restrict__ sq,
           int* __restrict__ knn) {
    extern __shared__ float smem[];
    float* xqT  = smem;                       // [CH][16]
    float* xc   = smem + CH * 16;             // [16][CH][16]
    float* dist = xc + 16 * CH * 16;          // [16][NPTS]

    const int b    = blockIdx.x / (NPTS / 16);
    const int mblk = blockIdx.x % (NPTS / 16);
    const int m0   = mblk * 16;
    const int tid  = threadIdx.x;
    const float* xb = x + (size_t)b * CH * NPTS;

    // --- async stage of query tile: xqT[c][m] = x[b][c][m0+m] ---------------
    {
        const int c = tid >> 2, part = tid & 3;          // 512 B128 transfers
        async_g2l_b128(lds_addr(&xqT[c * 16 + part * 4]),
                       xb + (size_t)c * NPTS + m0 + part * 4);
    }
    wait_async0();
    __syncthreads();

    const int wave = tid >> 5;
    const int lane = tid & 31;
    const int half = lane >> 4;     // K pair select / M-row half select
    const int ln   = lane & 15;
    float* xcw = xc + wave * (CH * 16);

    // Preload all 32 A-operand pairs (query tile) into registers: reused by
    // every column tile of this wave.  64 VGPRs.
    v2f areg[32];
#pragma unroll
    for (int kk = 0; kk < 32; ++kk) {
        areg[kk].x = xqT[(4 * kk + 2 * half)     * 16 + ln];
        areg[kk].y = xqT[(4 * kk + 2 * half + 1) * 16 + ln];
    }

    // --- WMMA distance tiles -------------------------------------------------
    for (int t = 0; t < 8; ++t) {
        const int n0 = wave * 128 + t * 16;

        // async stage candidate tile: xcw[c][j] = x[b][c][n0+j]
        for (int it = 0; it < 16; ++it) {
            const int idx = lane + 32 * it;              // 0..511
            const int c = idx >> 2, part = idx & 3;
            async_g2l_b128(lds_addr(&xcw[c * 16 + part * 4]),
                           xb + (size_t)c * NPTS + n0 + part * 4);
        }
        wait_async0();   // wave-private buffer: no cross-wave barrier needed

        v8f acc = {};
        // B pipeline: groups of 8 K-steps, double register bank, with hard
        // scheduling fences so the load block stays ahead of the WMMA block.
        v2f breg[8], bnxt[8];
#pragma unroll
        for (int u = 0; u < 8; ++u) {                    // group 0 prologue
            breg[u].x = xcw[(4 * u + 2 * half)     * 16 + ln];
            breg[u].y = xcw[(4 * u + 2 * half + 1) * 16 + ln];
        }
#pragma unroll
        for (int g = 0; g < 4; ++g) {
            const int kb2 = ((g + 1) & 3) * 32;          // next group base k
#pragma unroll
            for (int u = 0; u < 8; ++u) {                // 8 loads in flight
                bnxt[u].x = xcw[(kb2 + 4 * u + 2 * half)     * 16 + ln];
                bnxt[u].y = xcw[(kb2 + 4 * u + 2 * half + 1) * 16 + ln];
            }
            sched_fence();                               // loads stay above
#pragma unroll
            for (int u = 0; u < 8; ++u)                  // 8 WMMAs overlap them
                acc = wmma_f32(areg[g * 8 + u], breg[u], acc);
            sched_fence();                               // WMMAs stay above
#pragma unroll
            for (int u = 0; u < 8; ++u) breg[u] = bnxt[u];
        }

        // dist = |q|^2 + |c|^2 - 2*inner
        const float sqn = sq[(size_t)b * NPTS + n0 + ln];
        for (int r = 0; r < 8; ++r) {
            int m = half * 8 + r;
            float d = sq[(size_t)b * NPTS + m0 + m] + sqn - 2.0f * acc[r];
            dist[m * NPTS + n0 + ln] = d;
        }
    }
    __syncthreads();

    // --- top-32 smallest per row (wave `row` owns row `row`) ----------------
    const int row = wave;
    float* drow = dist + row * NPTS;
    for (int s = 0; s < KNN; ++s) {
        float bv = __builtin_inff();
        int   bi = NPTS;                       // sentinel > any index
        for (int j = lane; j < NPTS; j += 32) {
            float v = drow[j];
            if (v < bv || (v == bv && j < bi)) { bv = v; bi = j; }
        }
        for (int off = 16; off > 0; off >>= 1) {
            float ov = __shfl_down(bv, off, 32);
            int   oi = __shfl_down(bi, off, 32);
            if (ov < bv || (ov == bv && oi < bi)) { bv = ov; bi = oi; }
        }
        bi = __shfl(bi, 0, 32);
        if (lane == 0) {
            knn[((size_t)b * NPTS + m0 + row) * KNN + s] = bi;
            drow[bi] = __builtin_inff();       // knock out winner
        }
        __syncthreads();                       // LDS write visible, uniform
    }
}

// ---------------------------------------------------------------------------
// Kernel 3: per point n — gather-diff neighborhood, K=Wk*nb, V=Wv*nb via
// fp32 WMMA, q=Wq*x_n GEMV, local softmax attention, std score, o = attn*V.
// One workgroup (256 thr = 8 waves) per (b, n).
// A (global / LOADcnt) and B (LDS / DScnt) operands pipelined in groups of 8
// with sched_barrier(0) fences.  __launch_bounds__(256,1) for VGPR headroom.
// ---------------------------------------------------------------------------
__global__ void __launch_bounds__(256, 1)
point_kernel(const float* __restrict__ x,
             const float* __restrict__ Wq,
             const float* __restrict__ Wk,
             const float* __restrict__ Wv,
             const int*   __restrict__ knn,
             float* __restrict__ score,   // (B,H,N)
             float* __restrict__ oph) {   // (B,H,N,DEPTH)
    __shared__ float nb[CH * KNN];      // [c][j]   neighbor-diff
    __shared__ float kb[CH * KNN];      // [o][j]   K features
    __shared__ float vb[CH * KNN];      // [o][j]   V features
    __shared__ float qv[CH];
    __shared__ float attn_s[HEADS * KNN];
    __shared__ int   knl[KNN];

    const int b = blockIdx.x / NPTS;
    const int n = blockIdx.x % NPTS;
    const int tid = threadIdx.x;
    const float* xb = x + (size_t)b * CH * NPTS;

    if (tid < KNN) knl[tid] = knn[((size_t)b * NPTS + n) * KNN + tid];
    __syncthreads();

    // gather-diff: nb[c][j] = x[b][c][knn_j] - x[b][c][n]
    for (int i = tid; i < CH * KNN; i += 256) {
        int c = i >> 5, j = i & 31;
        nb[i] = xb[(size_t)c * NPTS + knl[j]] - xb[(size_t)c * NPTS + n];
    }
    // q = Wq * x[:,n]
    if (tid < CH) {
        float s = 0.f;
        for (int c = 0; c < CH; ++c) s += Wq[tid * CH + c] * xb[(size_t)c * NPTS + n];
        qv[tid] = s;
    }
    __syncthreads();

    const int wave = tid >> 5;
    const int lane = tid & 31;
    const int half = lane >> 4;
    const int ln   = lane & 15;

    // 32 output tiles: which(K/V) x 8 M-tiles x 2 N-tiles; each wave does 4.
    for (int tt = 0; tt < 4; ++tt) {
        const int tile  = wave * 4 + tt;
        const int which = tile >> 4;            // 0 -> K, 1 -> V
        const int mt    = (tile & 15) >> 1;
        const int nt    = tile & 1;
        const float* W  = which ? Wv : Wk;
        float* outb     = which ? vb : kb;
        const int o0 = mt * 16, j0 = nt * 16;
        const float* Wrow = W + (size_t)(o0 + ln) * CH;

        v8f acc = {};
        v2f aR[8], aN[8], bR[8], bN[8];
#pragma unroll
        for (int u = 0; u < 8; ++u) {                    // group 0 prologue
            aR[u].x = Wrow[4 * u + 2 * half];
            aR[u].y = Wrow[4 * u + 2 * half + 1];
            bR[u].x = nb[(4 * u + 2 * half)     * KNN + j0 + ln];
            bR[u].y = nb[(4 * u + 2 * half + 1) * KNN + j0 + ln];
        }
#pragma unroll
        for (int g = 0; g < 4; ++g) {
            const int kb2 = ((g + 1) & 3) * 32;          // next group base k
#pragma unroll
            for (int u = 0; u < 8; ++u) {
                aN[u].x = Wrow[kb2 + 4 * u + 2 * half];
                aN[u].y = Wrow[kb2 + 4 * u + 2 * half + 1];
                bN[u].x = nb[(kb2 + 4 * u + 2 * half)     * KNN + j0 + ln];
                bN[u].y = nb[(kb2 + 4 * u + 2 * half + 1) * KNN + j0 + ln];
            }
            sched_fence();                               // loads stay above
#pragma unroll
            for (int u = 0; u < 8; ++u)
                acc = wmma_f32(aR[u], bR[u], acc);
            sched_fence();                               // WMMAs stay above
#pragma unroll
            for (int u = 0; u < 8; ++u) { aR[u] = aN[u]; bR[u] = bN[u]; }
        }
        for (int r = 0; r < 8; ++r)
            outb[(o0 + half * 8 + r) * KNN + j0 + ln] = acc[r];
    }
    __syncthreads();

    // attention: wave h (<4) handles head h; lane j = neighbor index
    if (wave < HEADS) {
        const int h = wave, j = lane;
        float e = 0.f;
        for (int d = 0; d < DEPTH; ++d)
            e += qv[h * DEPTH + d] * kb[(h * DEPTH + d) * KNN + j];
        e *= 0.17677669529663687f;                         // 1/sqrt(32)
        float mx = e;
        for (int off = 16; off > 0; off >>= 1)
            mx = fmaxf(mx, __shfl_xor(mx, off, 32));
        float ex = expf(e - mx);
        float sm = ex;
        for (int off = 16; off > 0; off >>= 1)
            sm += __shfl_xor(sm, off, 32);
        float a = ex / sm;
        attn_s[h * KNN + j] = a;
        // std over the 32 attn weights (mean is exactly 1/32)
        float s2 = a * a;
        for (int off = 16; off > 0; off >>= 1)
            s2 += __shfl_xor(s2, off, 32);
        if (lane == 0) {
            const float mean = 1.0f / 32.0f;
            float var = s2 * (1.0f / 32.0f) - mean * mean;
            score[((size_t)b * HEADS + h) * NPTS + n] = sqrtf(fmaxf(var, 0.f));
        }
    }
    __syncthreads();

    // o[h][d] = sum_j attn[h][j] * V[h*32+d][j]; wave h (<4), lane = d
    if (wave < HEADS) {
        const int h = wave, d = lane;
        float s = 0.f;
        for (int j = 0; j < KNN; ++j)
            s += attn_s[h * KNN + j] * vb[(h * DEPTH + d) * KNN + j];
        oph[(((size_t)b * HEADS + h) * NPTS + n) * DEPTH + d] = s;
    }
}

// ---------------------------------------------------------------------------
// Kernel 4: per (b,h) full bitonic sort of 2048 (score,index) packed keys.
// drop==0: order = descending score, ties smaller index first (top_k(score))
// drop==1: order = ascending  score, ties smaller index first (top_k(-score))
// Writes the first TOPK indices of the sorted order.
// ---------------------------------------------------------------------------
__global__ void sort_kernel(const float* __restrict__ score,
                            int* __restrict__ outIdx, int drop) {
    __shared__ unsigned long long keys[NPTS];
    const int bh  = blockIdx.x;
    const int tid = threadIdx.x;
    const float* s = score + (size_t)bh * NPTS;

    for (int i = tid; i < NPTS; i += 1024) {
        unsigned u = __float_as_uint(s[i]);
        u = (u & 0x80000000u) ? ~u : (u | 0x80000000u);   // monotone ascending
        if (!drop) u = ~u;                                // descending by score
        keys[i] = ((unsigned long long)u << 32) | (unsigned)i;
    }
    __syncthreads();

    for (int ksz = 2; ksz <= NPTS; ksz <<= 1) {
        for (int j = ksz >> 1; j > 0; j >>= 1) {
            for (int i = tid; i < NPTS; i += 1024) {
                int ixj = i ^ j;
                if (ixj > i) {
                    bool up = ((i & ksz) == 0);           // ascending segment
                    unsigned long long a = keys[i], c = keys[ixj];
                    if ((a > c) == up) { keys[i] = c; keys[ixj] = a; }
                }
            }
            __syncthreads();
        }
    }
    for (int i = tid; i < TOPK; i += 1024)
        outIdx[(size_t)bh * TOPK + i] = (int)(keys[i] & 0xffffffffu);
}

// ---------------------------------------------------------------------------
// Kernel 5: emit v_down (B,128,1024), idx (B,4,1024) as float,
//                v_drop (B,128,1024), idx_drop (B,4,1024) as float.
// One block per (b,m); 128 threads = output channels (h*32+d).
// ---------------------------------------------------------------------------
__global__ void emit_kernel(const float* __restrict__ oph,
                            const int* __restrict__ idxKeep,
                            const int* __restrict__ idxDrop,
                            float* __restrict__ out) {
    const size_t OFF_VDOWN = 0;
    const size_t OFF_IDX   = (size_t)BATCH * CH * TOPK;                 // 1048576
    const size_t OFF_VDROP = OFF_IDX + (size_t)BATCH * HEADS * TOPK;    // +32768
    const size_t OFF_IDXD  = OFF_VDROP + (size_t)BATCH * CH * TOPK;

    const int b = blockIdx.x >> 10;
    const int m = blockIdx.x & 1023;
    const int t = threadIdx.x;             // channel = h*32 + d
    const int h = t >> 5, d = t & 31;

    const int pk = idxKeep[((size_t)b * HEADS + h) * TOPK + m];
    out[OFF_VDOWN + ((size_t)b * CH + t) * TOPK + m] =
        oph[(((size_t)b * HEADS + h) * NPTS + pk) * DEPTH + d];

    const int pd = idxDrop[((size_t)b * HEADS + h) * TOPK + m];
    out[OFF_VDROP + ((size_t)b * CH + t) * TOPK + m] =
        oph[(((size_t)b * HEADS + h) * NPTS + pd) * DEPTH + d];

    if (t < HEADS) {
        out[OFF_IDX + ((size_t)b * HEADS + t) * TOPK + m] =
            (float)idxKeep[((size_t)b * HEADS + t) * TOPK + m];
    } else if (t < 2 * HEADS) {
        const int h2 = t - HEADS;
        out[OFF_IDXD + ((size_t)b * HEADS + h2) * TOPK + m] =
            (float)idxDrop[((size_t)b * HEADS + h2) * TOPK + m];
    }
}

// ---------------------------------------------------------------------------
// Host launcher
// ---------------------------------------------------------------------------
extern "C" void kernel_launch(void* const* d_in, const int* in_sizes, int n_in,
                              void* d_out, int out_size, void* d_ws, size_t ws_size,
                              hipStream_t stream) {
    const float* x  = (const float*)d_in[0];
    const float* Wq = (const float*)d_in[1];
    const float* Wk = (const float*)d_in[2];
    const float* Wv = (const float*)d_in[3];
    float* out = (float*)d_out;

    // workspace layout (byte offsets)
    char* ws = (char*)d_ws;
    float* sq      = (float*)(ws);                                   //  64 KB
    int*   knn     = (int*)  (ws + 65536);                           //   2 MB
    float* score   = (float*)(ws + 65536 + 2097152);                 // 256 KB
    float* oph     = (float*)(ws + 65536 + 2097152 + 262144);        //   8 MB
    int*   idxKeep = (int*)  (ws + 65536 + 2097152 + 262144 + 8388608);
    int*   idxDrop = (int*)  (ws + 65536 + 2097152 + 262144 + 8388608 + 131072);

    // 1) squared norms
    sq_kernel<<<(BATCH * NPTS) / 256, 256, 0, stream>>>(x, sq);

    // 2) WMMA distances + top-32 KNN
    //    dyn LDS = xqT (CH*16) + xc (16*CH*16) + dist (16*NPTS) floats = 264 KB
    const size_t knn_lds =
        (size_t)(CH * 16 + 16 * CH * 16 + 16 * NPTS) * sizeof(float);
    knn_kernel<<<BATCH * (NPTS / 16), 512, knn_lds, stream>>>(x, sq, knn);

    // 3) gather + WMMA K/V GEMMs + attention + score
    point_kernel<<<BATCH * NPTS, 256, 0, stream>>>(x, Wq, Wk, Wv, knn, score, oph);

    // 4) top-1024 keep / drop orders per (b,h)
    sort_kernel<<<BATCH * HEADS, 1024, 0, stream>>>(score, idxKeep, 0);
    sort_kernel<<<BATCH * HEADS, 1024, 0, stream>>>(score, idxDrop, 1);

    // 5) gather outputs
    emit_kernel<<<BATCH * TOPK, 128, 0, stream>>>(oph, idxKeep, idxDrop, out);
}